// Lfm2MoeAttention_4982162063770
// MI455X (gfx1250) — compile-verified
//
#include <hip/hip_runtime.h>
#include <hip/hip_bf16.h>
#include <math.h>

typedef float v2f __attribute__((ext_vector_type(2)));
typedef float v8f __attribute__((ext_vector_type(8)));

static __device__ __forceinline__ v8f wmma4(v2f a, v2f b, v8f c) {
  // D = A(16x4 f32) * B(4x16 f32) + C(16x16 f32)
  return __builtin_amdgcn_wmma_f32_16x16x4_f32(false, a, false, b, (short)0, c,
                                               false, false);
}

// Issue one 16-byte global -> LDS async copy (per-lane), tracked by ASYNCcnt.
static __device__ __forceinline__ void async_g2l_b128(const float* gptr,
                                                      float* lptr) {
  const unsigned long long ga = (unsigned long long)(uintptr_t)gptr;
  // addrspacecast(LDS->generic) keeps the LDS byte offset in the low 32 bits,
  // which is exactly what the DS unit consumes.
  const unsigned la = (unsigned)(uintptr_t)lptr;
  asm volatile("global_load_async_to_lds_b128 %0, %1, off"
               :
               : "v"(la), "v"(ga)
               : "memory");
}

// ---------------------------------------------------------------------------
// GEMM: C[M,N] = A[M,K] @ B[K,N], fp32 row-major.
// Block = 256 threads (8 waves as 2x4). Block tile 64x128, K-step 32,
// double-buffered LDS staging via GLOBAL_LOAD_ASYNC_TO_LDS_B128 (6 async
// ops per thread per tile; s_wait_asynccnt 6 after issuing the next tile
// guarantees the current tile is resident since async loads retire in order).
// Each wave computes a 32x32 C region = 2x2 WMMA tiles (4 accumulators), so
// each kk step issues 2 A-frag + 2 B-frag LDS loads for 4 WMMAs.
// Requires M % 64 == 0, N % 128 == 0, K % 32 == 0 (true for all calls).
// ---------------------------------------------------------------------------
__global__ __launch_bounds__(256) void gemm_f32_wmma(
    const float* __restrict__ A, const float* __restrict__ B,
    float* __restrict__ C, int M, int N, int K) {
  __shared__ float As[2][64][36];   // [buf][Mtile][Ktile] (+pad)
  __shared__ float Bs[2][32][132];  // [buf][Ktile][Ntile] (+pad)

  const int tid  = threadIdx.x;
  const int lane = tid & 31;
  const int wave = tid >> 5;
  const int wm   = wave >> 2;  // 0..1 -> 32-row slab
  const int wn   = wave & 3;   // 0..3 -> 32-col slab

  const int m0 = blockIdx.x * 64;
  const int n0 = blockIdx.y * 128;

  const int fm = lane & 15;          // row-within-tile (A) / col (B)
  const int kb = (lane >> 4) << 1;   // 0 or 2
  const int ro = (lane >> 4) << 3;   // 0 or 8 (C row offset)

  v8f acc00 = {}, acc01 = {}, acc10 = {}, acc11 = {};

  const int arow  = tid >> 3;         // 0..31 (and +32)
  const int acol4 = (tid & 7) << 2;   // 0..28
  const int brow  = tid >> 3;         // 0..31
  const int bcol4 = (tid & 7) << 2;   // 0..28 (then +32*j)

  // Stage tile k0 of A (64x32) and B (32x128) into LDS buffer `buf`:
  // 6 async b128 ops per thread.
  auto stage = [&](int buf, int k0) {
    async_g2l_b128(&A[(size_t)(m0 + arow) * K + k0 + acol4],
                   &As[buf][arow][acol4]);
    async_g2l_b128(&A[(size_t)(m0 + arow + 32) * K + k0 + acol4],
                   &As[buf][arow + 32][acol4]);
#pragma unroll
    for (int j = 0; j < 4; ++j)
      async_g2l_b128(&B[(size_t)(k0 + brow) * N + n0 + bcol4 + 32 * j],
                     &Bs[buf][brow][bcol4 + 32 * j]);
  };

  stage(0, 0);

  const int ntiles = K >> 5;
  for (int i = 0; i < ntiles; ++i) {
    const int cur = i & 1;
    if (i + 1 < ntiles) {
      stage(cur ^ 1, (i + 1) << 5);
      // current tile's 6 asyncs were issued before these 6 => done at cnt<=6
      asm volatile("s_wait_asynccnt 0x6" ::: "memory");
    } else {
      asm volatile("s_wait_asynccnt 0x0" ::: "memory");
    }
    __syncthreads();

#pragma unroll
    for (int kk = 0; kk < 32; kk += 4) {
      v2f a0, a1, b0, b1;
      a0.x = As[cur][wm * 32 + fm][kk + kb];
      a0.y = As[cur][wm * 32 + fm][kk + kb + 1];
      a1.x = As[cur][wm * 32 + 16 + fm][kk + kb];
      a1.y = As[cur][wm * 32 + 16 + fm][kk + kb + 1];
      b0.x = Bs[cur][kk + kb][wn * 32 + fm];
      b0.y = Bs[cur][kk + kb + 1][wn * 32 + fm];
      b1.x = Bs[cur][kk + kb][wn * 32 + 16 + fm];
      b1.y = Bs[cur][kk + kb + 1][wn * 32 + 16 + fm];
      acc00 = wmma4(a0, b0, acc00);
      acc01 = wmma4(a0, b1, acc01);
      acc10 = wmma4(a1, b0, acc10);
      acc11 = wmma4(a1, b1, acc11);
    }
    __syncthreads();
  }

  const int rbase = m0 + wm * 32;
  const int cbase = n0 + wn * 32;
#pragma unroll
  for (int r = 0; r < 8; ++r) {
    C[(size_t)(rbase + r + ro) * N + cbase + fm]           = acc00[r];
    C[(size_t)(rbase + r + ro) * N + cbase + 16 + fm]      = acc01[r];
    C[(size_t)(rbase + 16 + r + ro) * N + cbase + fm]      = acc10[r];
    C[(size_t)(rbase + 16 + r + ro) * N + cbase + 16 + fm] = acc11[r];
  }
}

// ---------------------------------------------------------------------------
// Per-head RMSNorm + RoPE on the qkv buffer (T x 3072).
// One wave per (token, head); lane handles dims d=lane and d=lane+32 which is
// exactly the RoPE (x1, x2) pairing for D=64.
// ---------------------------------------------------------------------------
__global__ __launch_bounds__(32) void norm_rope_kernel(
    float* __restrict__ qkv, const int* __restrict__ positions,
    const float* __restrict__ q_ln_w, const float* __restrict__ k_ln_w) {
  const int t    = blockIdx.x;
  const int h    = blockIdx.y;  // 0..31 => Q head, 32..39 => K head
  const int lane = threadIdx.x;

  const bool isQ = (h < 32);
  float* base = qkv + (size_t)t * 3072 + (isQ ? h * 64 : 2048 + (h - 32) * 64);
  const float* w = isQ ? q_ln_w : k_ln_w;

  float x1 = base[lane];
  float x2 = base[lane + 32];

  float ss = x1 * x1 + x2 * x2;
#pragma unroll
  for (int off = 16; off >= 1; off >>= 1) ss += __shfl_xor(ss, off, 32);
  const float inv = rsqrtf(ss * (1.0f / 64.0f) + 1e-5f);

  x1 = x1 * inv * w[lane];
  x2 = x2 * inv * w[lane + 32];

  const float pos  = (float)positions[t];
  const float freq = powf(1000000.0f, -(float)lane / 32.0f);  // theta^(-2i/D)
  const float ang  = pos * freq;
  float s, c;
  __sincosf(ang, &s, &c);

  base[lane]      = x1 * c - x2 * s;
  base[lane + 32] = x2 * c + x1 * s;
}

// ---------------------------------------------------------------------------
// Causal GQA flash attention. One wave per (32-query tile, q-head).
// Two 16-row query subtiles share every K / V B-fragment load, so each
// global load feeds 2 WMMAs. Online softmax with 16-lane shuffles; the
// P tiles are routed through LDS to convert C-layout -> A-layout.
// Fully-masked trailing key tiles for the upper subtile are numerically
// benign (p=0, corr=1), so the ragged diagonal needs no special casing.
// ---------------------------------------------------------------------------
__global__ __launch_bounds__(32) void attn_kernel(
    const float* __restrict__ qkv, float* __restrict__ attn) {
  __shared__ float Plds[32][17];

  const int q0   = blockIdx.x * 32;
  const int nq   = blockIdx.y;   // query head 0..31
  const int kv   = nq >> 2;      // G = 4
  const int lane = threadIdx.x;
  const int n    = lane & 15;
  const int kb   = (lane >> 4) << 1;
  const int ro   = (lane >> 4) << 3;

  const int LD = 3072;
  const float* Qb = qkv + nq * 64;
  const float* Kb = qkv + 2048 + kv * 64;
  const float* Vb = qkv + 2048 + 512 + kv * 64;

  // Load Q tile (32x64) as 2 x 16 A-fragments.
  v2f aq[2][16];
#pragma unroll
  for (int i = 0; i < 2; ++i)
#pragma unroll
    for (int c = 0; c < 16; ++c) {
      const float* p = Qb + (size_t)(q0 + i * 16 + n) * LD + 4 * c + kb;
      aq[i][c].x = p[0];
      aq[i][c].y = p[1];
    }

  float m_i[2][8], l_i[2][8];
  v8f o[2][4];
#pragma unroll
  for (int i = 0; i < 2; ++i) {
#pragma unroll
    for (int r = 0; r < 8; ++r) { m_i[i][r] = -1e30f; l_i[i][r] = 0.0f; }
#pragma unroll
    for (int t4 = 0; t4 < 4; ++t4) o[i][t4] = (v8f){};
  }

  const float scale = 0.125f;  // 1/sqrt(64)

  for (int j0 = 0; j0 <= q0 + 16; j0 += 16) {
    // ---- S_i = Q_i K^T (two 16x16 tiles, shared K fragments) ----
    v8f s0 = {}, s1 = {};
#pragma unroll
    for (int c = 0; c < 16; ++c) {
      const float* p = Kb + (size_t)(j0 + n) * LD + 4 * c + kb;
      v2f b;
      b.x = p[0];
      b.y = p[1];
      s0 = wmma4(aq[0][c], b, s0);
      s1 = wmma4(aq[1][c], b, s1);
    }

    // ---- mask + online softmax (both subtiles) ----
    float pr[2][8];
#pragma unroll
    for (int i = 0; i < 2; ++i) {
#pragma unroll
      for (int r = 0; r < 8; ++r) {
        float sv = (i == 0 ? s0[r] : s1[r]) * scale;
        const int qi = q0 + i * 16 + r + ro;
        const int kj = j0 + n;
        if (kj > qi) sv = -1e30f;

        float mx = sv;
#pragma unroll
        for (int off = 8; off >= 1; off >>= 1)
          mx = fmaxf(mx, __shfl_xor(mx, off, 16));
        const float mnew = fmaxf(m_i[i][r], mx);

        const float pv = __expf(sv - mnew);
        float rs = pv;
#pragma unroll
        for (int off = 8; off >= 1; off >>= 1) rs += __shfl_xor(rs, off, 16);

        const float corr = __expf(m_i[i][r] - mnew);
        l_i[i][r] = l_i[i][r] * corr + rs;
        m_i[i][r] = mnew;
        pr[i][r]  = pv;

        o[i][0][r] *= corr; o[i][1][r] *= corr;
        o[i][2][r] *= corr; o[i][3][r] *= corr;
      }
    }

    // ---- C-layout P -> LDS -> A-layout fragments ----
#pragma unroll
    for (int i = 0; i < 2; ++i)
#pragma unroll
      for (int r = 0; r < 8; ++r) Plds[i * 16 + r + ro][n] = pr[i][r];
    __syncthreads();
    v2f pa[2][4];
#pragma unroll
    for (int i = 0; i < 2; ++i)
#pragma unroll
      for (int c = 0; c < 4; ++c) {
        pa[i][c].x = Plds[i * 16 + n][4 * c + kb];
        pa[i][c].y = Plds[i * 16 + n][4 * c + kb + 1];
      }
    __syncthreads();

    // ---- O_i += P_i V (shared V fragments) ----
#pragma unroll
    for (int c = 0; c < 4; ++c) {
#pragma unroll
      for (int t4 = 0; t4 < 4; ++t4) {
        const float* p = Vb + (size_t)(j0 + 4 * c + kb) * LD + t4 * 16 + n;
        v2f b;
        b.x = p[0];
        b.y = p[LD];  // next key row
        o[0][t4] = wmma4(pa[0][c], b, o[0][t4]);
        o[1][t4] = wmma4(pa[1][c], b, o[1][t4]);
      }
    }
  }

  // ---- normalize and store (T x 2048, head-major like the reference) ----
#pragma unroll
  for (int i = 0; i < 2; ++i)
#pragma unroll
    for (int r = 0; r < 8; ++r) {
      const float inv = 1.0f / l_i[i][r];
      const int row = q0 + i * 16 + r + ro;
#pragma unroll
      for (int t4 = 0; t4 < 4; ++t4)
        attn[(size_t)row * 2048 + nq * 64 + t4 * 16 + n] = o[i][t4][r] * inv;
    }
}

// ---------------------------------------------------------------------------
extern "C" void kernel_launch(void* const* d_in, const int* in_sizes, int n_in,
                              void* d_out, int out_size, void* d_ws,
                              size_t ws_size, hipStream_t stream) {
  const int*   positions = (const int*)d_in[0];
  const float* hidden    = (const float*)d_in[1];
  const float* w_qkv     = (const float*)d_in[2];
  const float* w_out     = (const float*)d_in[3];
  const float* q_ln_w    = (const float*)d_in[4];
  const float* k_ln_w    = (const float*)d_in[5];
  float*       out       = (float*)d_out;

  const int T   = in_sizes[0];  // 2048
  const int HID = 2048;
  const int QKV = 3072;         // (32 + 2*8) * 64

  float* qkv  = (float*)d_ws;                  // T x 3072
  float* attn = qkv + (size_t)T * QKV;         // T x 2048

  // 1) qkv = hidden @ w_qkv
  dim3 g1(T / 64, QKV / 128);
  gemm_f32_wmma<<<g1, 256, 0, stream>>>(hidden, w_qkv, qkv, T, QKV, HID);

  // 2) per-head RMSNorm + RoPE on q,k
  dim3 g2(T, 40);
  norm_rope_kernel<<<g2, 32, 0, stream>>>(qkv, positions, q_ln_w, k_ln_w);

  // 3) causal GQA attention (32 queries per wave)
  dim3 g3(T / 32, 32);
  attn_kernel<<<g3, 32, 0, stream>>>(qkv, attn);

  // 4) out = attn @ w_out
  dim3 g4(T / 64, HID / 128);
  gemm_f32_wmma<<<g4, 256, 0, stream>>>(attn, w_out, out, T, HID, HID);
}